// GCN_Body_bn_84275848282320
// MI455X (gfx1250) — compile-verified
//
#include <hip/hip_runtime.h>
#include <hip/hip_bf16.h>

#define GCN_N 50000
#define GCN_E 800000
#define GCN_H 256
#define BN_EPS 1e-5f

typedef __attribute__((ext_vector_type(2))) float v2f;
typedef __attribute__((ext_vector_type(8))) float v8f;

// ---------------- normalization precompute ----------------

__global__ void deg_init_kernel(float* deg, int n) {
    int i = blockIdx.x * blockDim.x + threadIdx.x;
    if (i < n) deg[i] = 1.0f;  // self-loop contributes 1
}

__global__ void deg_accum_kernel(const int* __restrict__ dst, float* deg, int e) {
    int i = blockIdx.x * blockDim.x + threadIdx.x;
    if (i < e) atomicAdd(&deg[dst[i]], 1.0f);
}

__global__ void dinv_kernel(float* deg, int n) {
    int i = blockIdx.x * blockDim.x + threadIdx.x;
    if (i < n) deg[i] = rsqrtf(deg[i]);  // in place: deg -> d^-1/2
}

__global__ void norm_kernel(const int* __restrict__ src, const int* __restrict__ dst,
                            const float* __restrict__ dinv, float* __restrict__ norm, int e) {
    int i = blockIdx.x * blockDim.x + threadIdx.x;
    if (i < e) norm[i] = dinv[src[i]] * dinv[dst[i]];
}

// ---------------- dense transform: C[M,256] = A[M,256] @ B[256,256] ----------------
// One wave owns a 16x64 strip of C: 4 accumulators, A fragment reused 4x.
// f32 WMMA (16x16x4) keeps full fp32 fidelity with the reference; GEMM is
// ~20 GFLOP total and nowhere near the bottleneck, so no precision tradeoff needed.

__global__ __launch_bounds__(128) void gemm_wmma_f32(const float* __restrict__ A,
                                                     const float* __restrict__ B,
                                                     float* __restrict__ C, int M) {
    const int K = 256, NC = 256;
    const int wave = (int)((blockIdx.x * blockDim.x + threadIdx.x) >> 5);
    const int lane = threadIdx.x & 31;
    const int strips = M >> 4;                 // M/16 row strips
    const int rowStrip = wave % strips;
    const int colStrip = wave / strips;        // 0..3 (64 cols each)
    if (colStrip >= 4) return;

    const int m0 = rowStrip << 4;
    const int n0 = colStrip << 6;
    const int r16  = lane & 15;                // row (A) / col (B,C) within tile
    const int koff = (lane >> 4) << 1;         // 0 or 2: upper half-wave holds K+2,K+3

    v8f acc0 = {}, acc1 = {}, acc2 = {}, acc3 = {};
    const float* Arow = A + (size_t)(m0 + r16) * K;

    for (int k = 0; k < K; k += 4) {
        // A fragment: lane holds A[m0+r16, k+koff .. k+koff+1]
        v2f a = *(const v2f*)(Arow + k + koff);
        // B fragments: lane holds B[k+koff, n] , B[k+koff+1, n]
        const float* B0 = B + (size_t)(k + koff) * NC + n0 + r16;
        const float* B1 = B0 + NC;
        v2f b0 = { B0[0],  B1[0]  };
        v2f b1 = { B0[16], B1[16] };
        v2f b2 = { B0[32], B1[32] };
        v2f b3 = { B0[48], B1[48] };
        acc0 = __builtin_amdgcn_wmma_f32_16x16x4_f32(false, a, false, b0, (short)0, acc0, false, false);
        acc1 = __builtin_amdgcn_wmma_f32_16x16x4_f32(false, a, false, b1, (short)0, acc1, false, false);
        acc2 = __builtin_amdgcn_wmma_f32_16x16x4_f32(false, a, false, b2, (short)0, acc2, false, false);
        acc3 = __builtin_amdgcn_wmma_f32_16x16x4_f32(false, a, false, b3, (short)0, acc3, false, false);
    }

    // C layout: VGPR r -> M = m0 + r + (lane>=16 ? 8 : 0), N = n0 + (lane&15)
    const int cm = m0 + ((lane >> 4) << 3);
    float* Cp = C + (size_t)cm * NC + n0 + r16;
#pragma unroll
    for (int r = 0; r < 8; ++r) {
        Cp[(size_t)r * NC +  0] = acc0[r];
        Cp[(size_t)r * NC + 16] = acc1[r];
        Cp[(size_t)r * NC + 32] = acc2[r];
        Cp[(size_t)r * NC + 48] = acc3[r];
    }
}

// ---------------- aggregation ----------------
// out[i,:] = b + hw[i,:]*dinv[i]^2   (self-loop fused with bias init, no atomics)

__global__ void self_init_kernel(const float* __restrict__ hw, const float* __restrict__ bias,
                                 const float* __restrict__ dinv, float* __restrict__ out, int n) {
    long long idx = (long long)blockIdx.x * blockDim.x + threadIdx.x;
    if (idx >= (long long)n * 64) return;
    int i = (int)(idx >> 6);
    int c = (int)(idx & 63) << 2;
    float w = dinv[i]; w *= w;
    float4 v  = *(const float4*)(hw + (size_t)i * GCN_H + c);
    float4 bb = *(const float4*)(bias + c);
    float4 o  = { bb.x + v.x * w, bb.y + v.y * w, bb.z + v.z * w, bb.w + v.w * w };
    *(float4*)(out + (size_t)i * GCN_H + c) = o;
}

// out[dst[e],:] += hw[src[e],:]*norm[e]. h is 51 MB -> L2 resident; atomics hit L2.

__global__ void scatter_kernel(const float* __restrict__ hw, const int* __restrict__ src,
                               const int* __restrict__ dst, const float* __restrict__ norm,
                               float* out, int e) {
    long long idx = (long long)blockIdx.x * blockDim.x + threadIdx.x;
    if (idx >= (long long)e * 64) return;
    int ed = (int)(idx >> 6);
    int c  = (int)(idx & 63) << 2;
    int s = src[ed], d = dst[ed];
    float w = norm[ed];
    float4 v = *(const float4*)(hw + (size_t)s * GCN_H + c);
    float* o = out + (size_t)d * GCN_H + c;
    atomicAdd(o + 0, v.x * w);
    atomicAdd(o + 1, v.y * w);
    atomicAdd(o + 2, v.z * w);
    atomicAdd(o + 3, v.w * w);
}

// ---------------- BatchNorm (training mode, biased variance) ----------------

__global__ void zero_stats_kernel(float* stats) {
    int i = threadIdx.x;
    if (i < 2 * GCN_H) stats[i] = 0.0f;
}

__global__ __launch_bounds__(256) void bn_stats_kernel(const float* __restrict__ h,
                                                       float* stats, int n) {
    int f = threadIdx.x;  // one thread per feature -> fully coalesced row reads
    float s = 0.f, q = 0.f;
    for (int i = blockIdx.x; i < n; i += gridDim.x) {
        float v = h[(size_t)i * GCN_H + f];
        s += v;
        q += v * v;
    }
    atomicAdd(&stats[f], s);
    atomicAdd(&stats[GCN_H + f], q);
}

__global__ void bn_apply_kernel(const float* __restrict__ h, const float* __restrict__ stats,
                                const float* __restrict__ g, const float* __restrict__ be,
                                float* __restrict__ out, int n) {
    long long idx = (long long)blockIdx.x * blockDim.x + threadIdx.x;
    if (idx >= (long long)n * 64) return;
    int i = (int)(idx >> 6);
    int c = (int)(idx & 63) << 2;
    const float inv_n = 1.0f / (float)n;
    float4 v = *(const float4*)(h + (size_t)i * GCN_H + c);
    float4 o;
    {
        float m = stats[c + 0] * inv_n;
        float var = stats[GCN_H + c + 0] * inv_n - m * m;
        o.x = (v.x - m) * rsqrtf(var + BN_EPS) * g[c + 0] + be[c + 0];
    }
    {
        float m = stats[c + 1] * inv_n;
        float var = stats[GCN_H + c + 1] * inv_n - m * m;
        o.y = (v.y - m) * rsqrtf(var + BN_EPS) * g[c + 1] + be[c + 1];
    }
    {
        float m = stats[c + 2] * inv_n;
        float var = stats[GCN_H + c + 2] * inv_n - m * m;
        o.z = (v.z - m) * rsqrtf(var + BN_EPS) * g[c + 2] + be[c + 2];
    }
    {
        float m = stats[c + 3] * inv_n;
        float var = stats[GCN_H + c + 3] * inv_n - m * m;
        o.w = (v.w - m) * rsqrtf(var + BN_EPS) * g[c + 3] + be[c + 3];
    }
    *(float4*)(out + (size_t)i * GCN_H + c) = o;
}

// ---------------- driver ----------------

extern "C" void kernel_launch(void* const* d_in, const int* in_sizes, int n_in,
                              void* d_out, int out_size, void* d_ws, size_t ws_size,
                              hipStream_t stream) {
    (void)in_sizes; (void)n_in; (void)out_size; (void)ws_size;

    const float* x  = (const float*)d_in[0];
    const int*   ei = (const int*)d_in[1];
    const int* src = ei;
    const int* dst = ei + GCN_E;
    const float* W1 = (const float*)d_in[2];  const float* b1 = (const float*)d_in[3];
    const float* g1 = (const float*)d_in[4];  const float* be1 = (const float*)d_in[5];
    const float* W2 = (const float*)d_in[6];  const float* b2 = (const float*)d_in[7];
    const float* g2 = (const float*)d_in[8];  const float* be2 = (const float*)d_in[9];
    const float* W3 = (const float*)d_in[10]; const float* b3 = (const float*)d_in[11];
    const float* g3 = (const float*)d_in[12]; const float* be3 = (const float*)d_in[13];

    float* ws    = (float*)d_ws;
    float* dinv  = ws;                                   // [N]
    float* norm  = dinv + GCN_N;                         // [E]
    float* hw    = norm + GCN_E;                         // [N*H] gemm result
    float* agg   = hw + (size_t)GCN_N * GCN_H;           // [N*H] aggregation / hidden
    float* stats = agg + (size_t)GCN_N * GCN_H;          // [2*H]

    // D^-1/2 and per-edge norm (once per call)
    deg_init_kernel<<<(GCN_N + 255) / 256, 256, 0, stream>>>(dinv, GCN_N);
    deg_accum_kernel<<<(GCN_E + 255) / 256, 256, 0, stream>>>(dst, dinv, GCN_E);
    dinv_kernel<<<(GCN_N + 255) / 256, 256, 0, stream>>>(dinv, GCN_N);
    norm_kernel<<<(GCN_E + 255) / 256, 256, 0, stream>>>(src, dst, dinv, norm, GCN_E);

    const unsigned nodeBlocks = (unsigned)(((long long)GCN_N * 64 + 255) / 256);
    const unsigned edgeBlocks = (unsigned)(((long long)GCN_E * 64 + 255) / 256);
    const unsigned gemmBlocks = (GCN_N / 16) * 4 / 4;  // 12500 waves, 4 waves/block

    auto layer = [&](const float* hin, const float* W, const float* b,
                     const float* g, const float* be, float* outb) {
        gemm_wmma_f32<<<gemmBlocks, 128, 0, stream>>>(hin, W, hw, GCN_N);
        self_init_kernel<<<nodeBlocks, 256, 0, stream>>>(hw, b, dinv, outb, GCN_N);
        scatter_kernel<<<edgeBlocks, 256, 0, stream>>>(hw, src, dst, norm, outb, GCN_E);
        zero_stats_kernel<<<1, 512, 0, stream>>>(stats);
        bn_stats_kernel<<<1024, 256, 0, stream>>>(outb, stats, GCN_N);
        bn_apply_kernel<<<nodeBlocks, 256, 0, stream>>>(outb, stats, g, be, outb, GCN_N);
    };

    layer(x,   W1, b1, g1, be1, agg);
    layer(agg, W2, b2, g2, be2, agg);
    layer(agg, W3, b3, g3, be3, (float*)d_out);
}